// kmeans_67181878444634
// MI455X (gfx1250) — compile-verified
//
#include <hip/hip_runtime.h>
#include <hip/hip_bf16.h>

typedef __attribute__((ext_vector_type(2))) float v2f;
typedef __attribute__((ext_vector_type(8))) float v8f;

#define DDIM 512
#define KDIM 64
#define NBLOCKS 256
#define NTHREADS 256
#define PFROWS 32   // prefetch distance in rows (8 n-steps ahead)

// ---------------------------------------------------------------------------
// Zero the workspace accumulators (G partial buffer + hh scalar).
// ---------------------------------------------------------------------------
__global__ void zero_ws_kernel(float* __restrict__ ws, int n) {
    int i = blockIdx.x * blockDim.x + threadIdx.x;
    if (i < n) ws[i] = 0.0f;
}

// ---------------------------------------------------------------------------
// Fused pass over h / F:
//   - accumulates G = F^T h  (K=64 x D=512) via v_wmma_f32_16x16x4_f32
//   - accumulates hh = sum(h*h) from the already-loaded B-matrix registers
// Each block handles rows [blockIdx.x*chunk, ...) and the FULL 64x512 output;
// wave w owns the 64 contiguous d-columns [64w, 64w+64), all 4 k-tiles.
// Partials folded into global Gbuf with hardware f32 atomics (L2-resident).
// ---------------------------------------------------------------------------
__global__ void __launch_bounds__(NTHREADS)
gemm_partial_kernel(const float* __restrict__ h, const float* __restrict__ F,
                    float* __restrict__ Gbuf, float* __restrict__ hhsum,
                    int N, int chunk) {
    const int lane = threadIdx.x & 31;
    const int wave = threadIdx.x >> 5;
    const int m    = lane & 15;     // column / row-in-half index
    const int hi   = lane >> 4;     // 0: lanes 0-15, 1: lanes 16-31
    const int d0w  = wave * 64;     // this wave's 64 d-columns

    // Prefetch lane pattern: lane l covers row (+l>>3), 32-byte chunk (l&7):
    // one global_prefetch_b8 touches 4 rows x 256B = the wave's full n-step
    // working set of h (8 cachelines), resp. 4 full rows of F.
    const int pfRow = lane >> 3;          // 0..3
    const int pfCol = (lane & 7) * 8;     // 8-float (32B) stride

    const int n0 = blockIdx.x * chunk;
    int nEnd = n0 + chunk;
    if (nEnd > N) nEnd = N;

    v8f acc[4][4];                  // [k-tile][d-tile] 16x16 f32 accumulators
#pragma unroll
    for (int kt = 0; kt < 4; ++kt)
#pragma unroll
        for (int dt = 0; dt < 4; ++dt)
            acc[kt][dt] = (v8f){0.f, 0.f, 0.f, 0.f, 0.f, 0.f, 0.f, 0.f};

    float hh = 0.0f;

#pragma unroll 2
    for (int n = n0; n < nEnd; n += 4) {
        // ---- A = F^T tile rows n..n+3 (16x4 f32 layout):
        //      VGPR0: K=0 (lanes 0-15) / K=2 (lanes 16-31); VGPR1: K=1 / K=3
        const float* Fr0 = F + (size_t)(n + 2 * hi) * KDIM;
        const float* Fr1 = Fr0 + KDIM;
        v2f A[4];
#pragma unroll
        for (int kt = 0; kt < 4; ++kt) {
            A[kt].x = Fr0[kt * 16 + m];
            A[kt].y = Fr1[kt * 16 + m];
        }

        // ---- B = h tile (4x16 f32), mirrored layout; fuse hh += b*b
        const float* Hr0 = h + (size_t)(n + 2 * hi) * DDIM;
        const float* Hr1 = Hr0 + DDIM;
        v2f B[4];
#pragma unroll
        for (int dt = 0; dt < 4; ++dt) {
            float b0 = Hr0[d0w + dt * 16 + m];
            float b1 = Hr1[d0w + dt * 16 + m];
            B[dt].x = b0;
            B[dt].y = b1;
            hh = fmaf(b0, b0, hh);
            hh = fmaf(b1, b1, hh);
        }

        // ---- WGP-scope prefetch PFROWS ahead (scope 0 = fill all levels).
        // WGP-scope prefetches are non-speculative, so clamp the row to stay
        // inside the allocation (scalar min: n is wave-uniform).
        {
            int pn = n + PFROWS;
            if (pn > N - 4) pn = N - 4;
            const float* pfH = h + (size_t)(pn + pfRow) * DDIM + d0w + pfCol;
            const float* pfF = F + (size_t)(pn + pfRow) * KDIM + pfCol;
            asm volatile("global_prefetch_b8 %0, off" :: "v"(pfH));
            asm volatile("global_prefetch_b8 %0, off" :: "v"(pfF));
        }

        // ---- 16 independent WMMAs: no acc RAW hazard within a step
#pragma unroll
        for (int kt = 0; kt < 4; ++kt)
#pragma unroll
            for (int dt = 0; dt < 4; ++dt)
                acc[kt][dt] = __builtin_amdgcn_wmma_f32_16x16x4_f32(
                    /*neg_a=*/false, A[kt], /*neg_b=*/false, B[dt],
                    /*c_mod=*/(short)0, acc[kt][dt],
                    /*reuse_a=*/false, /*reuse_b=*/false);
    }

    // ---- fold partial G into global buffer (L2-resident f32 atomics)
    // C/D layout: VGPR v, lane l -> row v + 8*(l>=16), col l%16
#pragma unroll
    for (int kt = 0; kt < 4; ++kt)
#pragma unroll
        for (int dt = 0; dt < 4; ++dt)
#pragma unroll
            for (int v = 0; v < 8; ++v) {
                const int gk = kt * 16 + v + 8 * hi;
                const int gd = d0w + dt * 16 + m;
                __hip_atomic_fetch_add(&Gbuf[gk * DDIM + gd], acc[kt][dt][v],
                                       __ATOMIC_RELAXED, __HIP_MEMORY_SCOPE_AGENT);
            }

    // ---- hh: wave32 butterfly reduce, one atomic per wave
#pragma unroll
    for (int off = 16; off > 0; off >>= 1) hh += __shfl_xor(hh, off, 32);
    if (lane == 0)
        __hip_atomic_fetch_add(hhsum, hh, __ATOMIC_RELAXED, __HIP_MEMORY_SCOPE_AGENT);
}

// ---------------------------------------------------------------------------
// out[0] = hh - sum(G*G)   (G is 64x512 = 32768 floats)
// ---------------------------------------------------------------------------
__global__ void __launch_bounds__(NTHREADS)
finalize_kernel(const float* __restrict__ Gbuf, const float* __restrict__ hhsum,
                float* __restrict__ out) {
    __shared__ float red[NTHREADS / 32];
    float s = 0.0f;
    for (int i = threadIdx.x; i < KDIM * DDIM; i += NTHREADS) {
        float g = Gbuf[i];
        s = fmaf(g, g, s);
    }
#pragma unroll
    for (int off = 16; off > 0; off >>= 1) s += __shfl_xor(s, off, 32);
    const int lane = threadIdx.x & 31, wave = threadIdx.x >> 5;
    if (lane == 0) red[wave] = s;
    __syncthreads();
    if (threadIdx.x == 0) {
        float t = 0.0f;
#pragma unroll
        for (int w = 0; w < NTHREADS / 32; ++w) t += red[w];
        out[0] = hhsum[0] - t;
    }
}

// ---------------------------------------------------------------------------
extern "C" void kernel_launch(void* const* d_in, const int* in_sizes, int n_in,
                              void* d_out, int out_size, void* d_ws, size_t ws_size,
                              hipStream_t stream) {
    const float* h = (const float*)d_in[0];   // [N, 512] f32
    const float* F = (const float*)d_in[1];   // [N, 64]  f32
    float* Gbuf  = (float*)d_ws;              // 64*512 partial G
    float* hhsum = Gbuf + KDIM * DDIM;        // 1 scalar
    float* out   = (float*)d_out;

    const int N     = in_sizes[0] / DDIM;
    const int chunk = (N + NBLOCKS - 1) / NBLOCKS;

    const int nzero = KDIM * DDIM + 1;
    zero_ws_kernel<<<(nzero + NTHREADS - 1) / NTHREADS, NTHREADS, 0, stream>>>(Gbuf, nzero);
    gemm_partial_kernel<<<NBLOCKS, NTHREADS, 0, stream>>>(h, F, Gbuf, hhsum, N, chunk);
    finalize_kernel<<<1, NTHREADS, 0, stream>>>(Gbuf, hhsum, out);
}